// NormalizedCirculantMask2_51685636440287
// MI455X (gfx1250) — compile-verified
//
#include <hip/hip_runtime.h>

typedef float v4f __attribute__((ext_vector_type(4)));

#define ROW_C   4096
#define THREADS 256
// Each thread moves 4 float4's per row: 256 threads * 4 elems * 4 passes = 4096.
#define PASSES  (ROW_C / (THREADS * 4))

__global__ __launch_bounds__(THREADS)
void NormalizedCirculantMask2_kernel(const float* __restrict__ x,
                                     const float* __restrict__ weight,
                                     float* __restrict__ out)
{
    __shared__ float smem[ROW_C];

    const int row = blockIdx.x;            // one 4096-float row per workgroup
    const int tid = threadIdx.x;
    const float* xrow = x  + (size_t)row * ROW_C;
    float*       orow = out + (size_t)row * ROW_C;

    // ---- Stage the full row (16 KB) into LDS via gfx1250 async copies ----
    // global_load_async_to_lds_b128: VDST = LDS byte address, VADDR = 64-bit
    // global address (GV mode, no SADDR). Tracked by ASYNCcnt.
#pragma unroll
    for (int p = 0; p < PASSES; ++p) {
        const unsigned elem = (unsigned)(p * (THREADS * 4) + tid * 4);
        // low 32 bits of a generic LDS pointer == wave-relative LDS byte offset
        const unsigned lds_byte = (unsigned)(uintptr_t)(&smem[elem]);
        const unsigned long long gaddr = (unsigned long long)(uintptr_t)(xrow + elem);
        asm volatile("global_load_async_to_lds_b128 %0, %1, off"
                     :
                     : "v"(lds_byte), "v"(gaddr)
                     : "memory");
    }
    // Drain this wave's async copies, then make all waves' data visible.
    asm volatile("s_wait_asynccnt 0x0" ::: "memory");
    __syncthreads();

    const float w = weight[0];
    const float u = 1.0f - w;

    // ---- out[i] = w*x[i] + (1-w)*x[(i+1) mod 4096], vectorized x4 ----
#pragma unroll
    for (int p = 0; p < PASSES; ++p) {
        const int i = p * (THREADS * 4) + tid * 4;
        v4f v = *(const v4f*)(&smem[i]);
        const float nxt = smem[(i + 4) & (ROW_C - 1)];   // circular wrap
        v4f r;
        r.x = w * v.x + u * v.y;
        r.y = w * v.y + u * v.z;
        r.z = w * v.z + u * v.w;
        r.w = w * v.w + u * nxt;
        // Streaming output (256 MB) would only thrash L2 -> non-temporal b128.
        __builtin_nontemporal_store(r, (v4f*)(orow + i));
    }
}

extern "C" void kernel_launch(void* const* d_in, const int* in_sizes, int n_in,
                              void* d_out, int out_size, void* d_ws, size_t ws_size,
                              hipStream_t stream)
{
    const float* x      = (const float*)d_in[0];   // (4, 4096, 4096) f32
    const float* weight = (const float*)d_in[1];   // (1,) f32
    float*       out    = (float*)d_out;

    const int rows = out_size / ROW_C;             // 4 * 4096 = 16384 rows
    NormalizedCirculantMask2_kernel<<<dim3(rows), dim3(THREADS), 0, stream>>>(x, weight, out);
}